// PairwiseScorer_81982335746216
// MI455X (gfx1250) — compile-verified
//
#include <hip/hip_runtime.h>

typedef float v2f __attribute__((ext_vector_type(2)));
typedef float v8f __attribute__((ext_vector_type(8)));

#define KDIM 1024
#define DDIM 128

// ---------------------------------------------------------------------------
// Step 1: fold the 3 blocks of W1 (384x128) into two 128x128 matrices:
//   Wa[e,o] = W1[e,o]       + W1[256+e,o]   (multiplies span_i)
//   Wb[e,o] = W1[128+e,o]   - W1[256+e,o]   (multiplies antecedent_j)
// ---------------------------------------------------------------------------
__global__ void PairwiseScorer_prep_weights(const float* __restrict__ W1,
                                            float* __restrict__ Wa,
                                            float* __restrict__ Wb) {
    int idx = blockIdx.x * blockDim.x + threadIdx.x;   // 0..16383
    int e = idx >> 7;
    int o = idx & 127;
    float wtop = W1[(e      ) * DDIM + o];
    float wmid = W1[(128 + e) * DDIM + o];
    float wbot = W1[(256 + e) * DDIM + o];
    Wa[idx] = wtop + wbot;
    Wb[idx] = wmid - wbot;
}

// ---------------------------------------------------------------------------
// Step 2: fp32 WMMA GEMM  Out[1024x128] = In[1024x128] @ W[128x128] (+ bias)
// One wave per 16x16 output tile, chaining 32x V_WMMA_F32_16X16X4_F32 over
// K=128. blockIdx.y == 0 -> A = span@Wa + b1 ; blockIdx.y == 1 -> B = ant@Wb.
//
// A-fragment (16x4 f32, ISA 7.12.2): lane L<16 holds row M=L, K={0,1} in the
// two VGPRs; lane L>=16 holds row M=L-16, K={2,3}.  B/C/D: row striped across
// lanes within a VGPR; C/D VGPR r -> row r (lanes 0-15) / row r+8 (lanes 16-31).
// ---------------------------------------------------------------------------
__global__ void __launch_bounds__(256)
PairwiseScorer_gemm_wmma(const float* __restrict__ In0, const float* __restrict__ Wt0,
                         float* __restrict__ Out0,
                         const float* __restrict__ In1, const float* __restrict__ Wt1,
                         float* __restrict__ Out1,
                         const float* __restrict__ b1) {
    const float* In;  const float* Wt;  float* Out;  const float* bias;
    if (blockIdx.y == 0) { In = In0; Wt = Wt0; Out = Out0; bias = b1; }
    else                 { In = In1; Wt = Wt1; Out = Out1; bias = nullptr; }

    const int wave  = threadIdx.x >> 5;
    const int lane  = threadIdx.x & 31;
    const int tile  = blockIdx.x * 8 + wave;   // 0..511
    const int mTile = tile >> 3;               // 64 tiles of M
    const int nTile = tile & 7;                // 8 tiles of N

    const int lhalf = lane >> 4;               // 0 | 1
    const int l15   = lane & 15;
    const int row   = mTile * 16 + l15;        // A row this lane supplies
    const int col   = nTile * 16 + l15;        // B/C/D column this lane owns
    const int klo   = lhalf * 2;               // K offset within a K=4 step

    const float* arow = In + row * DDIM;

    v8f c = {};
#pragma unroll
    for (int k0 = 0; k0 < DDIM; k0 += 4) {
        v2f a = *(const v2f*)(arow + k0 + klo);            // A[row, k0+klo .. +1]
        v2f b;
        b.x = Wt[(k0 + klo    ) * DDIM + col];             // B[k0+klo,   col]
        b.y = Wt[(k0 + klo + 1) * DDIM + col];             // B[k0+klo+1, col]
        c = __builtin_amdgcn_wmma_f32_16x16x4_f32(
                /*neg_a=*/false, a, /*neg_b=*/false, b,
                /*c_mod=*/(short)0, c, /*reuse_a=*/false, /*reuse_b=*/false);
    }

    const float bcol  = bias ? bias[col] : 0.0f;           // fold b1 into A
    const int   rbase = mTile * 16 + lhalf * 8;
#pragma unroll
    for (int r = 0; r < 8; ++r) {
        Out[(rbase + r) * DDIM + col] = c[r] + bcol;
    }
}

// ---------------------------------------------------------------------------
// Step 3: pairwise scores.  out[i,j] = (i>j) ? relu(A[i,:]+B[j,:])·W2 + b2 : 0
// 16x16 thread tile per block; A/B rows are L2/L0 resident (512 KB each).
// ---------------------------------------------------------------------------
__global__ void __launch_bounds__(256)
PairwiseScorer_pairwise(const float* __restrict__ A, const float* __restrict__ Bm,
                        const float* __restrict__ W2, const float* __restrict__ b2,
                        float* __restrict__ out) {
    const int j = blockIdx.x * 16 + (threadIdx.x & 15);
    const int i = blockIdx.y * 16 + (threadIdx.x >> 4);

    float score = 0.0f;
    if (i > j) {
        const float4* ap = (const float4*)(A  + i * DDIM);
        const float4* bp = (const float4*)(Bm + j * DDIM);
        const float4* wp = (const float4*)W2;
        float acc = 0.0f;
#pragma unroll 4
        for (int d4 = 0; d4 < DDIM / 4; ++d4) {
            float4 a = ap[d4];
            float4 b = bp[d4];
            float4 w = wp[d4];
            acc = fmaf(fmaxf(a.x + b.x, 0.0f), w.x, acc);
            acc = fmaf(fmaxf(a.y + b.y, 0.0f), w.y, acc);
            acc = fmaf(fmaxf(a.z + b.z, 0.0f), w.z, acc);
            acc = fmaf(fmaxf(a.w + b.w, 0.0f), w.w, acc);
        }
        score = acc + b2[0];
    }
    out[i * KDIM + j] = score;   // every element written (d_out is poisoned)
}

// ---------------------------------------------------------------------------
extern "C" void kernel_launch(void* const* d_in, const int* in_sizes, int n_in,
                              void* d_out, int out_size, void* d_ws, size_t ws_size,
                              hipStream_t stream) {
    const float* span = (const float*)d_in[0];   // [1024,128]
    const float* ant  = (const float*)d_in[1];   // [1024,128]
    const float* W1   = (const float*)d_in[2];   // [384,128]
    const float* b1   = (const float*)d_in[3];   // [128]
    const float* W2   = (const float*)d_in[4];   // [128,1]
    const float* b2   = (const float*)d_in[5];   // [1]
    float*       out  = (float*)d_out;           // [1024,1024]

    float* ws = (float*)d_ws;
    float* Wa = ws;                              // 128*128
    float* Wb = ws + 16384;                      // 128*128
    float* A  = ws + 32768;                      // 1024*128 (span@Wa + b1)
    float* Bm = ws + 32768 + 131072;             // 1024*128 (ant@Wb)

    PairwiseScorer_prep_weights<<<dim3(16384 / 256), 256, 0, stream>>>(W1, Wa, Wb);
    PairwiseScorer_gemm_wmma<<<dim3(64, 2), 256, 0, stream>>>(span, Wa, A, ant, Wb, Bm, b1);
    PairwiseScorer_pairwise<<<dim3(64, 64), 256, 0, stream>>>(A, Bm, W2, b2, out);
}